// GatingRouting_21406117003933
// MI455X (gfx1250) — compile-verified
//
#include <hip/hip_runtime.h>

typedef __attribute__((ext_vector_type(2))) float v2f;
typedef __attribute__((ext_vector_type(8))) float v8f;

#define MAX_O 150.0f

// One wave (32 lanes) processes 16 rows per WMMA tile.
//   D(16x16) = A(16x4) x B(4x16) + C  chained over 16 K-steps (K=64 total)
//   A = W2^T slice  : A[m][kk] = W2[4k+kk][m]   (m = channel, only m<3 nonzero)
//   B = h^T slice   : B[kk][n] = relu(o_norm[n]*W1[4k+kk] + b1[4k+kk])
//   C/D layout      : VGPR r, lanes 0-15 -> M=r, N=lane  => lane n holds all 3
//                     logits of row n in c[0],c[1],c[2].
__global__ void __launch_bounds__(256) gating_routing_wmma_kernel(
    const float* __restrict__ o_batch,
    const float* __restrict__ W1,
    const float* __restrict__ b1,
    const float* __restrict__ W2,
    const float* __restrict__ b2,
    float* __restrict__ out,
    int batch)
{
    const int lane = threadIdx.x & 31;
    const int m    = lane & 15;              // A: M (channel) / B: N (row-in-tile)
    const int koff = (lane >> 4) ? 2 : 0;    // f32 A/B layout: hi half-wave = K+2,K+3

    // ---- loop-invariant per-lane weight fragments (L2-resident, loaded once) ----
    float w1a[16], w1b[16], b1a[16], b1b[16];
    v2f   afrag[16];
    #pragma unroll
    for (int k = 0; k < 16; ++k) {
        const int j0 = 4 * k + koff;
        const int j1 = j0 + 1;
        w1a[k] = W1[j0];  b1a[k] = b1[j0];
        w1b[k] = W1[j1];  b1b[k] = b1[j1];
        // W2 is (64,3) row-major: W2[j][m] at W2[j*3 + m]; pad channels m>=3 with 0
        afrag[k].x = (m < 3) ? W2[j0 * 3 + m] : 0.0f;
        afrag[k].y = (m < 3) ? W2[j1 * 3 + m] : 0.0f;
    }
    const float bb0 = b2[0], bb1 = b2[1], bb2 = b2[2];

    const int waveId = (blockIdx.x * blockDim.x + threadIdx.x) >> 5;
    const int nWaves = (gridDim.x * blockDim.x) >> 5;
    const int nTiles = (batch + 15) >> 4;

    for (int tile = waveId; tile < nTiles; tile += nWaves) {
        int row = tile * 16 + m;
        int rowc = row < batch ? row : batch - 1;   // clamp keeps EXEC all-1s
        const float o     = o_batch[rowc];
        const float onorm = o * (1.0f / MAX_O);

        // accumulator pre-seeded with b2 (channels live in c[0..2])
        v8f c = {bb0, bb1, bb2, 0.0f, 0.0f, 0.0f, 0.0f, 0.0f};

        #pragma unroll
        for (int k = 0; k < 16; ++k) {
            v2f b;
            b.x = fmaxf(fmaf(onorm, w1a[k], b1a[k]), 0.0f);
            b.y = fmaxf(fmaf(onorm, w1b[k], b1b[k]), 0.0f);
            // 8 args: (neg_a, A, neg_b, B, c_mod, C, reuse_a, reuse_b)
            c = __builtin_amdgcn_wmma_f32_16x16x4_f32(
                    false, afrag[k], false, b, (short)0, c, false, false);
        }

        // lanes 0-15 own one full row each; lanes 16-31 hold pad channels 8..15
        if (lane < 16 && row < batch) {
            const float l0 = c[0], l1 = c[1], l2 = c[2];
            const float mx = fmaxf(l0, fmaxf(l1, l2));
            const float e0 = __expf(l0 - mx);
            const float e1 = __expf(l1 - mx);
            const float e2 = __expf(l2 - mx);
            const float inv = 1.0f / (e0 + e1 + e2);
            float a0 = e0 * inv, a1 = e1 * inv, a2 = e2 * inv;
            if (o >= 100.0f) {
                a0 = 0.0f; a1 = 0.0f; a2 = 1.0f;
            } else if (o <= 10.0f) {
                const float inv12 = 1.0f / (a0 + a1);
                a0 *= inv12; a1 *= inv12; a2 = 0.0f;
            }
            float* p = out + (size_t)row * 3;
            p[0] = a0; p[1] = a1; p[2] = a2;
        }
    }
}

extern "C" void kernel_launch(void* const* d_in, const int* in_sizes, int n_in,
                              void* d_out, int out_size, void* d_ws, size_t ws_size,
                              hipStream_t stream) {
    const float* o_batch = (const float*)d_in[0];
    const float* W1      = (const float*)d_in[1];
    const float* b1      = (const float*)d_in[2];
    const float* W2      = (const float*)d_in[3];
    const float* b2      = (const float*)d_in[4];
    float* out = (float*)d_out;
    const int batch = in_sizes[0];   // o_batch is (BATCH,1)

    const int threads = 256;                 // 8 wave32 per block
    int blocks = 2048;                       // grid-stride over 16-row tiles
    const int nTiles = (batch + 15) >> 4;
    const int maxBlocks = (nTiles * 32 + threads - 1) / threads;  // >=1 tile/wave
    if (blocks > maxBlocks && maxBlocks > 0) blocks = maxBlocks;

    gating_routing_wmma_kernel<<<blocks, threads, 0, stream>>>(
        o_batch, W1, b1, W2, b2, out, batch);
}